// Attention_86852828659931
// MI455X (gfx1250) — compile-verified
//
#include <hip/hip_runtime.h>
#include <hip/hip_bf16.h>
#include <math.h>

typedef __attribute__((ext_vector_type(2))) float v2f;
typedef __attribute__((ext_vector_type(8))) float v8f;

#define N_NODES 100000
#define EDGES   800000
#define DIM     128
#define HEADS   4

#define XS_STRIDE 132   // padded LDS row stride (floats): bank = 4*(lane%16)+c mod 64 -> conflict-free b64 loads

// ---------------------------------------------------------------------------
// One wave computes a 32x32 output tile (2x2 grid of 16x16 WMMA tiles) of
//   out[n, j] = (sum_d Xs[n,d] * W[j,d] + bias[j]) * scale
// 4 WMMAs per K-step fed by 2 LDS b64 + 2 global b64 fragment loads, with
// four independent accumulation chains to hide load latency.
// ---------------------------------------------------------------------------
__device__ __forceinline__ void gemm_tile_2x2(const float* __restrict__ Xs,
                                              const float* __restrict__ W,
                                              const float* __restrict__ bias,
                                              float* __restrict__ out, float scale,
                                              int node0, int n0,
                                              int l16, int half)
{
    const int col0 = n0 + l16;
    const int col1 = n0 + 16 + l16;

    // C/D layout: VGPR r holds row (r + 8*half), col = lane%16 -> bias broadcast
    const float bv0 = bias[col0];
    const float bv1 = bias[col1];
    v8f c00, c01, c10, c11;
    for (int r = 0; r < 8; ++r) { c00[r] = bv0; c01[r] = bv1; c10[r] = bv0; c11[r] = bv1; }

    // A layout 16x4: VGPR0 = K{0|2}, VGPR1 = K{1|3} split by lane half -> float2 @ kk + 2*half
    // B layout 4x16 mirrors it; B[k][n] = W[n][k] (x @ W^T) -> float2 from W row n
    const int t0 = half * 2;
    const float* w0 = W + (size_t)col0 * DIM;
    const float* w1 = W + (size_t)col1 * DIM;
    for (int kk = 0; kk < DIM; kk += 4) {
        v2f b0 = *(const v2f*)(w0 + kk + t0);
        v2f b1 = *(const v2f*)(w1 + kk + t0);
        v2f a0 = *(const v2f*)(&Xs[l16 * XS_STRIDE + kk + t0]);
        v2f a1 = *(const v2f*)(&Xs[(16 + l16) * XS_STRIDE + kk + t0]);
        c00 = __builtin_amdgcn_wmma_f32_16x16x4_f32(false, a0, false, b0, (short)0, c00, false, false);
        c01 = __builtin_amdgcn_wmma_f32_16x16x4_f32(false, a0, false, b1, (short)0, c01, false, false);
        c10 = __builtin_amdgcn_wmma_f32_16x16x4_f32(false, a1, false, b0, (short)0, c10, false, false);
        c11 = __builtin_amdgcn_wmma_f32_16x16x4_f32(false, a1, false, b1, (short)0, c11, false, false);
    }

    for (int r = 0; r < 8; ++r) {
        const int row0 = node0 + half * 8 + r;
        const int row1 = row0 + 16;
        out[(size_t)row0 * DIM + col0] = c00[r] * scale;
        out[(size_t)row0 * DIM + col1] = c01[r] * scale;
        out[(size_t)row1 * DIM + col0] = c10[r] * scale;
        out[(size_t)row1 * DIM + col1] = c11[r] * scale;
    }
}

// cooperative load of one 32x128 fp32 row tile into padded LDS (128 threads)
__device__ __forceinline__ void load_x_tile(float* __restrict__ Xs,
                                            const float* __restrict__ in,
                                            int node0, int tid)
{
    const float4* g = (const float4*)(in + (size_t)node0 * DIM);
    for (int i = 0; i < 8; ++i) {
        int j = i * 128 + tid;             // float4 index in tile (1024 total)
        int r = j >> 5;                    // row (32 float4 per row)
        int c = j & 31;
        float4 val = g[j];
        float* dst = &Xs[r * XS_STRIDE + c * 4];
        dst[0] = val.x; dst[1] = val.y; dst[2] = val.z; dst[3] = val.w;
    }
}

// ---------------------------------------------------------------------------
// Fused q/k/v projection: X tile staged once, three GEMMs against it.
// Block: 128 threads = 4 waves; wave w owns output cols [32w, 32w+32).
// ---------------------------------------------------------------------------
__global__ __launch_bounds__(128)
void qkv_proj(const float* __restrict__ x,
              const float* __restrict__ Wq, const float* __restrict__ bq, float* __restrict__ q,
              const float* __restrict__ Wk, const float* __restrict__ bk, float* __restrict__ k,
              const float* __restrict__ Wv, const float* __restrict__ bv, float* __restrict__ v)
{
    __shared__ float Xs[32 * XS_STRIDE];
    const int tid   = threadIdx.x;
    const int lane  = tid & 31;
    const int l16   = lane & 15;
    const int half  = lane >> 4;
    const int n0    = (tid >> 5) * 32;     // wave's output column base
    const int node0 = blockIdx.x * 32;

    load_x_tile(Xs, x, node0, tid);
    __syncthreads();

    gemm_tile_2x2(Xs, Wq, bq, q, 0.5f, node0, n0, l16, half);  // ref quirk: q /= sqrt(H)
    gemm_tile_2x2(Xs, Wk, bk, k, 1.0f, node0, n0, l16, half);
    gemm_tile_2x2(Xs, Wv, bv, v, 1.0f, node0, n0, l16, half);
}

// single-matrix variant for the output projection
__global__ __launch_bounds__(128)
void out_proj(const float* __restrict__ in, const float* __restrict__ W,
              const float* __restrict__ bias, float* __restrict__ out)
{
    __shared__ float Xs[32 * XS_STRIDE];
    const int tid   = threadIdx.x;
    const int lane  = tid & 31;
    const int l16   = lane & 15;
    const int half  = lane >> 4;
    const int n0    = (tid >> 5) * 32;
    const int node0 = blockIdx.x * 32;

    load_x_tile(Xs, in, node0, tid);
    __syncthreads();

    gemm_tile_2x2(Xs, W, bias, out, 1.0f, node0, n0, l16, half);
}

// ---------------------------------------------------------------------------
// SDDMM: one wave per edge. Flat dim j has head j%4, so a per-lane float4 at
// j = 4*lane .. 4*lane+3 carries exactly one product per head; a full-wave
// shfl_xor butterfly yields all 4 head scores at once.
// ---------------------------------------------------------------------------
__global__ __launch_bounds__(256)
void sddmm_edge_scores(const float* __restrict__ q, const float* __restrict__ k,
                       const int* __restrict__ row, const int* __restrict__ col,
                       float4* __restrict__ s)
{
    const int gtid = blockIdx.x * blockDim.x + threadIdx.x;
    const int e    = gtid >> 5;
    const int lane = gtid & 31;
    if (e >= EDGES) return;
    const int r = row[e];
    const int c = col[e];
    const float4 qv = *(const float4*)(q + (size_t)r * DIM + lane * 4);
    const float4 kv = *(const float4*)(k + (size_t)c * DIM + lane * 4);
    float4 p;
    p.x = qv.x * kv.x; p.y = qv.y * kv.y; p.z = qv.z * kv.z; p.w = qv.w * kv.w;
    for (int off = 16; off >= 1; off >>= 1) {
        p.x += __shfl_xor(p.x, off, 32);
        p.y += __shfl_xor(p.y, off, 32);
        p.z += __shfl_xor(p.z, off, 32);
        p.w += __shfl_xor(p.w, off, 32);
    }
    if (lane == 0) s[e] = p;
}

// ---------------------------------------------------------------------------
// Fused segment softmax + SpMM: one wave per destination node.
// row[] is sorted -> binary search the node's edge range, then
// max / sum-exp reductions over edges (lane-strided + butterfly), then a
// serial edge loop accumulating a[e,h] * v[col[e]] with lanes across the
// 128 feature dims (float4 per lane, components = heads 0..3).
// ---------------------------------------------------------------------------
__global__ __launch_bounds__(256)
void softmax_spmm(const float* __restrict__ vmat, const int* __restrict__ row,
                  const int* __restrict__ col, const float4* __restrict__ s,
                  float* __restrict__ y)
{
    const int gtid = blockIdx.x * blockDim.x + threadIdx.x;
    const int n    = gtid >> 5;
    const int lane = gtid & 31;
    if (n >= N_NODES) return;

    int lo = 0, hi = EDGES;
    while (lo < hi) { int mid = (lo + hi) >> 1; if (row[mid] < n) lo = mid + 1; else hi = mid; }
    const int start = lo;
    hi = EDGES;
    while (lo < hi) { int mid = (lo + hi) >> 1; if (row[mid] <= n) lo = mid + 1; else hi = mid; }
    const int end = lo;

    float4 acc = make_float4(0.f, 0.f, 0.f, 0.f);
    if (end > start) {
        // pass 1: per-head running max
        float4 m = make_float4(-3.402823466e38f, -3.402823466e38f,
                               -3.402823466e38f, -3.402823466e38f);
        for (int e = start + lane; e < end; e += 32) {
            float4 sv = s[e];
            m.x = fmaxf(m.x, sv.x); m.y = fmaxf(m.y, sv.y);
            m.z = fmaxf(m.z, sv.z); m.w = fmaxf(m.w, sv.w);
        }
        for (int off = 16; off >= 1; off >>= 1) {
            m.x = fmaxf(m.x, __shfl_xor(m.x, off, 32));
            m.y = fmaxf(m.y, __shfl_xor(m.y, off, 32));
            m.z = fmaxf(m.z, __shfl_xor(m.z, off, 32));
            m.w = fmaxf(m.w, __shfl_xor(m.w, off, 32));
        }
        // pass 2: per-head sum of exp
        float4 z = make_float4(0.f, 0.f, 0.f, 0.f);
        for (int e = start + lane; e < end; e += 32) {
            float4 sv = s[e];
            z.x += __expf(sv.x - m.x); z.y += __expf(sv.y - m.y);
            z.z += __expf(sv.z - m.z); z.w += __expf(sv.w - m.w);
        }
        for (int off = 16; off >= 1; off >>= 1) {
            z.x += __shfl_xor(z.x, off, 32);
            z.y += __shfl_xor(z.y, off, 32);
            z.z += __shfl_xor(z.z, off, 32);
            z.w += __shfl_xor(z.w, off, 32);
        }
        const float4 rz = make_float4(1.f / z.x, 1.f / z.y, 1.f / z.z, 1.f / z.w);
        // pass 3: y[n,:] += a[e,h] * v[col[e],:]
        for (int e = start; e < end; ++e) {
            float4 sv = s[e];
            float4 a;
            a.x = __expf(sv.x - m.x) * rz.x;
            a.y = __expf(sv.y - m.y) * rz.y;
            a.z = __expf(sv.z - m.z) * rz.z;
            a.w = __expf(sv.w - m.w) * rz.w;
            const int c = col[e];
            const float4 vv = *(const float4*)(vmat + (size_t)c * DIM + lane * 4);
            acc.x += a.x * vv.x; acc.y += a.y * vv.y;
            acc.z += a.z * vv.z; acc.w += a.w * vv.w;
        }
    }
    *(float4*)(y + (size_t)n * DIM + lane * 4) = acc;
}

// ---------------------------------------------------------------------------
extern "C" void kernel_launch(void* const* d_in, const int* in_sizes, int n_in,
                              void* d_out, int out_size, void* d_ws, size_t ws_size,
                              hipStream_t stream)
{
    const float* x   = (const float*)d_in[0];
    const int*   row = (const int*)  d_in[1];
    const int*   col = (const int*)  d_in[2];
    const float* Wq  = (const float*)d_in[3];
    const float* bq  = (const float*)d_in[4];
    const float* Wk  = (const float*)d_in[5];
    const float* bk  = (const float*)d_in[6];
    const float* Wv  = (const float*)d_in[7];
    const float* bv  = (const float*)d_in[8];
    const float* Wo  = (const float*)d_in[9];
    const float* bo  = (const float*)d_in[10];
    float* out = (float*)d_out;

    const size_t nd_bytes = (size_t)N_NODES * DIM * sizeof(float);   // 51.2 MB
    char* ws = (char*)d_ws;
    float*  q = (float*) ws;  ws += nd_bytes;
    float*  k = (float*) ws;  ws += nd_bytes;
    float*  v = (float*) ws;  ws += nd_bytes;
    float4* s = (float4*)ws;  ws += (size_t)EDGES * HEADS * sizeof(float); // 12.8 MB
    float*  y = (float*) ws;  // 51.2 MB

    // fused q/k/v projections (fp32 WMMA, 2x2 register tiling)
    qkv_proj<<<N_NODES / 32, 128, 0, stream>>>(x, Wq, bq, q, Wk, bk, k, Wv, bv, v);

    // per-edge, per-head scores (one wave per edge)
    sddmm_edge_scores<<<(EDGES * 32) / 256, 256, 0, stream>>>(q, k, row, col, s);

    // fused segment softmax + SpMM (one wave per node)
    softmax_spmm<<<(N_NODES * 32) / 256, 256, 0, stream>>>(v, row, col, s, y);

    // output projection
    out_proj<<<N_NODES / 32, 128, 0, stream>>>(y, Wo, bo, out);
}